// PromptLearner_84713934946305
// MI455X (gfx1250) — compile-verified
//
#include <hip/hip_runtime.h>
#include <hip/hip_bf16.h>

// ---------------------------------------------------------------------------
// Problem constants (match the reference)
// ---------------------------------------------------------------------------
#define BATCH    4
#define N_CLS    1000
#define SEQ_LEN  77
#define CTX_DIM  512
#define VIS_DIM  512
#define HID      32          // VIS_DIM/16
#define N_CTX_B  4
#define N_CTX_A  4

typedef __attribute__((ext_vector_type(2))) float v2f;
typedef __attribute__((ext_vector_type(4))) float v4f;
typedef __attribute__((ext_vector_type(8))) float v8f;

// ---------------------------------------------------------------------------
// Kernel 1: bias = relu(X @ W1 + b1) @ W2 + b2     (4x512 result -> d_ws)
// Done on the CDNA5 matrix path with V_WMMA_F32_16X16X4_F32.
//   A (16x4 f32) per-lane layout: m = lane&15, kh = lane>>4,
//     a.x = A[m][2*kh], a.y = A[m][2*kh+1]
//   B (4x16 f32): n = lane&15, b.x = B[2*kh][n], b.y = B[2*kh+1][n]
//   C/D (16x16 f32): VGPR r holds M = r + 8*(lane>>4), N = lane&15
// M=4 rows are padded to 16 with zeros via per-lane selects (no EXEC masking
// around the WMMA, which requires EXEC == all ones).
// ---------------------------------------------------------------------------
__global__ __launch_bounds__(128)
void PromptLearner_mlp_wmma(const float* __restrict__ X,   // 4 x 512
                            const float* __restrict__ W1,  // 512 x 32
                            const float* __restrict__ B1,  // 32
                            const float* __restrict__ W2,  // 32 x 512
                            const float* __restrict__ B2,  // 512
                            float* __restrict__ bias)      // 4 x 512 (out)
{
    __shared__ float hid[BATCH * HID];

    const int tid  = threadIdx.x;
    const int wave = tid >> 5;
    const int lane = tid & 31;
    const int m    = lane & 15;      // A-row within tile
    const int kh   = lane >> 4;      // K half (0: K=0,1   1: K=2,3)
    const int n    = lane & 15;      // B/C column within tile
    const int mC   = (m < BATCH) ? m : 0;   // clamped row for safe loads

    // ---- Stage 1: hidden(4x32) = relu(X @ W1 + b1); waves 0,1 take one
    //      16-wide N-tile each. K = 512 in steps of 4.
    if (wave < 2) {
        const int ntile = wave;          // columns [ntile*16, ntile*16+16)
        v8f acc = {};
        for (int k = 0; k < VIS_DIM; k += 4) {
            const int k0 = k + 2 * kh;
            v2f a, b;
            float ax = X[mC * VIS_DIM + k0];
            float ay = X[mC * VIS_DIM + k0 + 1];
            a.x = (m < BATCH) ? ax : 0.0f;
            a.y = (m < BATCH) ? ay : 0.0f;
            b.x = W1[k0 * HID + ntile * 16 + n];
            b.y = W1[(k0 + 1) * HID + ntile * 16 + n];
            acc = __builtin_amdgcn_wmma_f32_16x16x4_f32(
                false, a, false, b, (short)0, acc, false, false);
        }
        if (kh == 0) {                   // lanes 0-15 hold M = 0..7 in acc[0..7]
            const float bb = B1[ntile * 16 + n];
            #pragma unroll
            for (int r = 0; r < BATCH; ++r) {
                float v = acc[r] + bb;
                hid[r * HID + ntile * 16 + n] = (v > 0.0f) ? v : 0.0f;
            }
        }
    }
    __syncthreads();

    // ---- Stage 2: bias(4x512) = hidden(4x32) @ W2 + b2; 32 N-tiles of 16,
    //      K = 32 in steps of 4; tiles strided across the 4 waves.
    for (int nt = wave; nt < CTX_DIM / 16; nt += 4) {
        v8f acc = {};
        #pragma unroll
        for (int k = 0; k < HID; k += 4) {
            const int k0 = k + 2 * kh;
            v2f a, b;
            float ax = hid[mC * HID + k0];
            float ay = hid[mC * HID + k0 + 1];
            a.x = (m < BATCH) ? ax : 0.0f;
            a.y = (m < BATCH) ? ay : 0.0f;
            b.x = W2[k0 * CTX_DIM + nt * 16 + n];
            b.y = W2[(k0 + 1) * CTX_DIM + nt * 16 + n];
            acc = __builtin_amdgcn_wmma_f32_16x16x4_f32(
                false, a, false, b, (short)0, acc, false, false);
        }
        if (kh == 0) {
            const float bb = B2[nt * 16 + n];
            #pragma unroll
            for (int r = 0; r < BATCH; ++r)
                bias[r * CTX_DIM + nt * 16 + n] = acc[r] + bb;
        }
    }
}

// ---------------------------------------------------------------------------
// Kernel 2: the bandwidth kernel. One block per (class, pos); 128 threads,
// each owning one float4-granule of the 512-wide channel dim. The
// source-select branch is uniform per block. Output stores are non-temporal
// (631 MB, zero reuse) so the embedding read stream keeps the L2 to itself.
// ---------------------------------------------------------------------------
__global__ __launch_bounds__(128)
void PromptLearner_fill(const float* __restrict__ emb,    // 1000 x 77 x 512
                        const float* __restrict__ ctx_b,  // 4 x 512
                        const float* __restrict__ ctx_a,  // 4 x 512
                        const float* __restrict__ bias,   // 4 x 512 (from ws)
                        const int*   __restrict__ name_lens, // 1000
                        float* __restrict__ out)          // 4 x 1000 x 77 x 512
{
    const int blk = blockIdx.x;              // c * SEQ_LEN + pos
    const int c   = blk / SEQ_LEN;
    const int pos = blk - c * SEQ_LEN;
    const int d   = threadIdx.x * 4;         // channel offset (v4f granule)

    const size_t cls_stride = (size_t)SEQ_LEN * CTX_DIM;          // 39424
    const size_t b_stride   = (size_t)N_CLS * SEQ_LEN * CTX_DIM;  // per-batch
    float* o = out + (size_t)c * cls_stride + (size_t)pos * CTX_DIM + d;

    const int nl      = name_lens[c];
    const int start_a = 1 + N_CTX_B + nl;

    const float* src = nullptr;              // block-uniform decision
    if (pos >= 1 && pos < 1 + N_CTX_B) {
        src = ctx_b + (size_t)(pos - 1) * CTX_DIM + d;
    } else if (pos >= start_a && pos < start_a + N_CTX_A) {
        src = ctx_a + (size_t)(pos - start_a) * CTX_DIM + d;
    }

    if (src) {
        const v4f cv = *(const v4f*)src;
        #pragma unroll
        for (int b = 0; b < BATCH; ++b) {
            const v4f bb = *(const v4f*)(bias + b * CTX_DIM + d);
            v4f r = cv + bb;
            __builtin_nontemporal_store(r, (v4f*)(o + (size_t)b * b_stride));
        }
    } else {
        const v4f ev =
            *(const v4f*)(emb + (size_t)c * cls_stride + (size_t)pos * CTX_DIM + d);
        #pragma unroll
        for (int b = 0; b < BATCH; ++b)
            __builtin_nontemporal_store(ev, (v4f*)(o + (size_t)b * b_stride));
    }
}

// ---------------------------------------------------------------------------
// Launcher
// ---------------------------------------------------------------------------
extern "C" void kernel_launch(void* const* d_in, const int* in_sizes, int n_in,
                              void* d_out, int out_size, void* d_ws, size_t ws_size,
                              hipStream_t stream) {
    const float* X    = (const float*)d_in[0];   // im_features 4x512
    const float* W1   = (const float*)d_in[1];   // 512x32
    const float* B1v  = (const float*)d_in[2];   // 32
    const float* W2   = (const float*)d_in[3];   // 32x512
    const float* B2v  = (const float*)d_in[4];   // 512
    const float* ctxB = (const float*)d_in[5];   // 4x512
    const float* ctxA = (const float*)d_in[6];   // 4x512
    const float* emb  = (const float*)d_in[7];   // 1000x77x512
    const int*   nl   = (const int*)d_in[8];     // 1000
    float*       out  = (float*)d_out;
    float*       bias = (float*)d_ws;            // 4x512 scratch (8 KB)

    PromptLearner_mlp_wmma<<<1, 128, 0, stream>>>(X, W1, B1v, W2, B2v, bias);
    PromptLearner_fill<<<N_CLS * SEQ_LEN, 128, 0, stream>>>(emb, ctxB, ctxA,
                                                            bias, nl, out);
}